// CGCNN_36395552866975
// MI455X (gfx1250) — compile-verified
//
#include <hip/hip_runtime.h>
#include <math.h>

#define BB      8
#define NN      20000
#define EE      320000
#define EMB     8
#define CENTERS 10
#define H1D     24
#define H2D     24
#define NBLK    6
#define CATD    24

#define TILES_PER_WAVE 10          // 160 edges per wave
#define WAVES_PER_BLOCK 8          // 256 threads
#define EDGE_BLOCKS_X  (EE / (16 * TILES_PER_WAVE) / WAVES_PER_BLOCK)   // 250

typedef __attribute__((ext_vector_type(2))) float v2f;
typedef __attribute__((ext_vector_type(8))) float v8f;

static __device__ __forceinline__ v2f mk2(float x, float y) { v2f r; r.x = x; r.y = y; return r; }

// ---------------------------------------------------------------------------
// Build fused weight matrix Wcat[i] (24 x 16): cols 0..7 = W_sig, 8..15 = W_sm
// ---------------------------------------------------------------------------
__global__ __launch_bounds__(256) void wcat_kernel(const float* __restrict__ Wsig,
                                                   const float* __restrict__ Wsm,
                                                   float* __restrict__ Wcat) {
    int t = blockIdx.x * 256 + threadIdx.x;
    if (t >= NBLK * CATD * 16) return;
    int i   = t / (CATD * 16);
    int rem = t % (CATD * 16);
    int k   = rem / 16;
    int n   = rem % 16;
    float v = (n < EMB) ? Wsig[(i * CATD + k) * EMB + n]
                        : Wsm [(i * CATD + k) * EMB + (n - EMB)];
    Wcat[t] = v;
}

// ---------------------------------------------------------------------------
// s[b,n,:] = sites[b,n] * W_site + b_site
// ---------------------------------------------------------------------------
__global__ __launch_bounds__(256) void s_init_kernel(const float* __restrict__ sites,
                                                     const float* __restrict__ Wsite,
                                                     const float* __restrict__ bsite,
                                                     float* __restrict__ s) {
    int t = blockIdx.x * 256 + threadIdx.x;
    if (t >= BB * NN) return;
    float x = sites[t];
    float o[EMB];
#pragma unroll
    for (int j = 0; j < EMB; ++j) o[j] = fmaf(x, Wsite[j], bsite[j]);
    float4* p = (float4*)(s + (size_t)t * EMB);
    p[0] = make_float4(o[0], o[1], o[2], o[3]);
    p[1] = make_float4(o[4], o[5], o[6], o[7]);
}

// ---------------------------------------------------------------------------
// bnd[b,e,:] = gaussian(bonds[b,e]) @ W_bond + b_bond   (done once, L2-resident)
// ---------------------------------------------------------------------------
__global__ __launch_bounds__(256) void bnd_kernel(const float* __restrict__ bonds,
                                                  const float* __restrict__ Wbond,
                                                  const float* __restrict__ bbond,
                                                  float* __restrict__ bnd) {
    int t = blockIdx.x * 256 + threadIdx.x;
    if (t >= BB * EE) return;
    float x = bonds[t];
    float o[EMB];
#pragma unroll
    for (int j = 0; j < EMB; ++j) o[j] = bbond[j];
#pragma unroll
    for (int c = 0; c < CENTERS; ++c) {
        float ctr = (float)c * (10.0f / 9.0f);   // linspace(0,10,10)
        float d   = x - ctr;
        float g   = __expf(-d * d);              // WIDTH = 1
#pragma unroll
        for (int j = 0; j < EMB; ++j) o[j] = fmaf(g, Wbond[c * EMB + j], o[j]);
    }
    float4* p = (float4*)(bnd + (size_t)t * EMB);
    p[0] = make_float4(o[0], o[1], o[2], o[3]);
    p[1] = make_float4(o[4], o[5], o[6], o[7]);
}

// ---------------------------------------------------------------------------
// Edge kernel: one wave handles TILES_PER_WAVE tiles of 16 edges.
// A = v(16x24), B = Wcat(24x16), chained V_WMMA_F32_16X16X4_F32 over K.
// blockIdx.y = batch index -> per-batch base pointers are block-uniform;
// all gather/scatter addresses are uniform_base + zext(u32 byte offset)
// to encourage SADDR-form vector memory ops.
// ---------------------------------------------------------------------------
__global__ __launch_bounds__(256) void edge_kernel(const float* __restrict__ s_cur,
                                                   float* __restrict__ s_next,
                                                   const float* __restrict__ bnd,
                                                   const int* __restrict__ idx1,
                                                   const int* __restrict__ idx2,
                                                   const float* __restrict__ Wcat_i,
                                                   const float* __restrict__ bsig_i,
                                                   const float* __restrict__ bsm_i) {
    const int lane = threadIdx.x & 31;
    const int wave = threadIdx.x >> 5;
    const int n    = lane & 15;   // column (output feature) / row within tile
    const int h    = lane >> 4;   // K-half selector per ISA A/B layout

    const int b      = blockIdx.y;                                  // block-uniform
    const int e_base = (blockIdx.x * WAVES_PER_BLOCK + wave) * (16 * TILES_PER_WAVE);

    // B-matrix fragments (held for all tiles): lane = column n, K rows 4c+2h, 4c+2h+1
    v2f w[6];
#pragma unroll
    for (int c = 0; c < 6; ++c) {
        const float* wr = Wcat_i + (4 * c + 2 * h) * 16 + n;
        w[c] = mk2(wr[0], wr[16]);
    }
    const float bs = bsig_i[n & 7];
    const float bm = bsm_i[n & 7];

    // Block-uniform per-batch bases (stay in SGPRs)
    const char* sB = (const char*)(s_cur  + (size_t)b * NN * EMB);
    char*       sN = (char*)      (s_next + (size_t)b * NN * EMB);
    const char* bB = (const char*)(bnd    + (size_t)b * EE * EMB);

    const unsigned hbyte = (unsigned)(8 * h);   // byte offset of lane-half inside 32B row
    const unsigned n4    = (unsigned)(n << 2);  // feature byte offset for scatter

    for (int t = 0; t < TILES_PER_WAVE; ++t) {
        const int edge = e_base + t * 16 + n;
        const int i1 = idx1[edge];
        const int i2 = idx2[edge];

        const unsigned off1 = ((unsigned)i1 << 5) + hbyte;
        const unsigned off2 = ((unsigned)i2 << 5) + hbyte;
        const unsigned offb = ((unsigned)edge << 5) + hbyte;
        const float2* p1 = (const float2*)(sB + off1);
        const float2* p2 = (const float2*)(sB + off2);
        const float2* pb = (const float2*)(bB + offb);
        float2 a0 = p1[0], a1 = p1[2];   // v[4c+2h], v[4c+2h+1] for c=0,1
        float2 a2 = p2[0], a3 = p2[2];   // c=2,3
        float2 a4 = pb[0], a5 = pb[2];   // c=4,5

        v8f acc = {};
        acc = __builtin_amdgcn_wmma_f32_16x16x4_f32(false, mk2(a0.x, a0.y), false, w[0], (short)0, acc, false, false);
        acc = __builtin_amdgcn_wmma_f32_16x16x4_f32(false, mk2(a1.x, a1.y), false, w[1], (short)0, acc, false, false);
        acc = __builtin_amdgcn_wmma_f32_16x16x4_f32(false, mk2(a2.x, a2.y), false, w[2], (short)0, acc, false, false);
        acc = __builtin_amdgcn_wmma_f32_16x16x4_f32(false, mk2(a3.x, a3.y), false, w[3], (short)0, acc, false, false);
        acc = __builtin_amdgcn_wmma_f32_16x16x4_f32(false, mk2(a4.x, a4.y), false, w[4], (short)0, acc, false, false);
        acc = __builtin_amdgcn_wmma_f32_16x16x4_f32(false, mk2(a5.x, a5.y), false, w[5], (short)0, acc, false, false);

        // Pre-scaled scatter byte offset, shuffled instead of the raw index.
        const int i1b = (int)((unsigned)i1 << 5);

        // D layout: lanes 0-15 VGPR r -> edge M=r; lanes 16-31 -> M=r+8; column N=lane&15.
        // cols 0..7 = sig preact, cols 8..15 = sm preact -> pair via shfl_xor(8).
#pragma unroll
        for (int r = 0; r < 8; ++r) {
            float d    = acc[r];
            float o    = __shfl_xor(d, 8, 32);                    // partner column n^8
            int   m    = r + 8 * h;                               // edge row in tile
            int   dstb = __shfl(i1b, m, 32);                      // byte offset of dst row
            // fast sigmoid: v_exp_f32 + v_rcp_f32 (TRANS ops, co-issue with VALU)
            float sig = __builtin_amdgcn_rcpf(1.0f + __expf(-(d + bs)));
            float val = sig * fmaxf(o + bm, 0.0f);
            if (n < 8) {
                atomicAdd((float*)(sN + ((unsigned)dstb + n4)), val);
            }
        }
    }
}

// ---------------------------------------------------------------------------
// vec[b,:] = mean_n s[b,n,:]
// ---------------------------------------------------------------------------
__global__ __launch_bounds__(256) void reduce_kernel(const float* __restrict__ s,
                                                     float* __restrict__ vec) {
    __shared__ float red[256][EMB];
    const int b   = blockIdx.x;
    const int tid = threadIdx.x;
    const float* sb = s + (size_t)b * NN * EMB;
    float acc[EMB];
#pragma unroll
    for (int j = 0; j < EMB; ++j) acc[j] = 0.0f;
    for (int nn = tid; nn < NN; nn += 256) {
        const float4* p = (const float4*)(sb + (size_t)nn * EMB);
        float4 a = p[0], c = p[1];
        acc[0] += a.x; acc[1] += a.y; acc[2] += a.z; acc[3] += a.w;
        acc[4] += c.x; acc[5] += c.y; acc[6] += c.z; acc[7] += c.w;
    }
#pragma unroll
    for (int j = 0; j < EMB; ++j) red[tid][j] = acc[j];
    __syncthreads();
    for (int st = 128; st > 0; st >>= 1) {
        if (tid < st) {
#pragma unroll
            for (int j = 0; j < EMB; ++j) red[tid][j] += red[tid + st][j];
        }
        __syncthreads();
    }
    if (tid < EMB) vec[b * EMB + tid] = red[0][tid] * (1.0f / (float)NN);
}

// ---------------------------------------------------------------------------
// Tiny readout MLP: 8 -> 24 -> 24 -> 1 (one lane per batch entry)
// ---------------------------------------------------------------------------
__global__ void mlp_kernel(const float* __restrict__ vec,
                           const float* __restrict__ W1, const float* __restrict__ b1,
                           const float* __restrict__ W2, const float* __restrict__ b2,
                           const float* __restrict__ W3, const float* __restrict__ b3,
                           float* __restrict__ out) {
    int b = threadIdx.x;
    if (b >= BB) return;
    float v[EMB];
#pragma unroll
    for (int j = 0; j < EMB; ++j) v[j] = vec[b * EMB + j];
    float hA[H1D];
    for (int k = 0; k < H1D; ++k) {
        float a = b1[k];
#pragma unroll
        for (int j = 0; j < EMB; ++j) a = fmaf(v[j], W1[j * H1D + k], a);
        hA[k] = fmaxf(a, 0.0f);
    }
    float hB[H2D];
    for (int k = 0; k < H2D; ++k) {
        float a = b2[k];
        for (int j = 0; j < H1D; ++j) a = fmaf(hA[j], W2[j * H2D + k], a);
        hB[k] = fmaxf(a, 0.0f);
    }
    float o = b3[0];
    for (int k = 0; k < H2D; ++k) o = fmaf(hB[k], W3[k], o);
    out[b] = o;
}

// ---------------------------------------------------------------------------
extern "C" void kernel_launch(void* const* d_in, const int* in_sizes, int n_in,
                              void* d_out, int out_size, void* d_ws, size_t ws_size,
                              hipStream_t stream) {
    const float* sites = (const float*)d_in[0];
    const float* bonds = (const float*)d_in[1];
    const int*   idx1  = (const int*)d_in[2];
    const int*   idx2  = (const int*)d_in[3];
    const float* Wsite = (const float*)d_in[4];
    const float* bsite = (const float*)d_in[5];
    const float* Wbond = (const float*)d_in[6];
    const float* bbond = (const float*)d_in[7];
    const float* Wsig  = (const float*)d_in[8];
    const float* bsig  = (const float*)d_in[9];
    const float* Wsm   = (const float*)d_in[10];
    const float* bsm   = (const float*)d_in[11];
    const float* W1    = (const float*)d_in[12];
    const float* b1    = (const float*)d_in[13];
    const float* W2    = (const float*)d_in[14];
    const float* b2    = (const float*)d_in[15];
    const float* W3    = (const float*)d_in[16];
    const float* b3    = (const float*)d_in[17];

    float* ws   = (float*)d_ws;
    float* sA   = ws;                                   // B*N*EMB
    float* sBuf = sA   + (size_t)BB * NN * EMB;         // B*N*EMB
    float* bndB = sBuf + (size_t)BB * NN * EMB;         // B*E*EMB
    float* Wcat = bndB + (size_t)BB * EE * EMB;         // 6*24*16
    float* vec  = Wcat + NBLK * CATD * 16;              // B*EMB

    wcat_kernel<<<(NBLK * CATD * 16 + 255) / 256, 256, 0, stream>>>(Wsig, Wsm, Wcat);
    s_init_kernel<<<(BB * NN + 255) / 256, 256, 0, stream>>>(sites, Wsite, bsite, sA);
    bnd_kernel<<<(BB * EE + 255) / 256, 256, 0, stream>>>(bonds, Wbond, bbond, bndB);

    float* cur = sA;
    float* nxt = sBuf;
    dim3 egrid(EDGE_BLOCKS_X, BB, 1);                   // (250, 8)
    for (int i = 0; i < NBLK; ++i) {
        hipMemcpyAsync(nxt, cur, (size_t)BB * NN * EMB * sizeof(float),
                       hipMemcpyDeviceToDevice, stream);
        edge_kernel<<<egrid, 256, 0, stream>>>(cur, nxt, bndB, idx1, idx2,
                                               Wcat + i * CATD * 16,
                                               bsig + i * EMB, bsm + i * EMB);
        float* tmp = cur; cur = nxt; nxt = tmp;
    }

    reduce_kernel<<<BB, 256, 0, stream>>>(cur, vec);
    mlp_kernel<<<1, 32, 0, stream>>>(vec, W1, b1, W2, b2, W3, b3, (float*)d_out);
}